// DGLGCN_2714419331422
// MI455X (gfx1250) — compile-verified
//
#include <hip/hip_runtime.h>
#include <hip/hip_bf16.h>

typedef __attribute__((ext_vector_type(2))) float v2f;
typedef __attribute__((ext_vector_type(4))) float v4f;
typedef __attribute__((ext_vector_type(8))) float v8f;

#define KDIM 128

static __device__ __forceinline__ v8f wmma_step(float sx, float ax, float ay,
                                                v2f b, v8f acc) {
    v2f a;
    a.x = ax * sx;
    a.y = ay * sx;
    return __builtin_amdgcn_wmma_f32_16x16x4_f32(false, a, false, b, (short)0,
                                                 acc, false, false);
}

// ---------------------------------------------------------------------------
// Degree accumulation: one thread per edge, float atomics (L2-resident, 800KB)
// ---------------------------------------------------------------------------
__global__ void deg_kernel(const long long* __restrict__ ei, int E,
                           float* __restrict__ deg_out, float* __restrict__ deg_in) {
    int e = blockIdx.x * blockDim.x + threadIdx.x;
    if (e >= E) return;
    int s = (int)ei[e];
    int d = (int)ei[(size_t)E + e];
    atomicAdd(&deg_out[s], 1.0f);
    atomicAdd(&deg_in[d], 1.0f);
}

__global__ void isqrt_kernel(float* __restrict__ deg, int n) {
    int i = blockIdx.x * blockDim.x + threadIdx.x;
    if (i >= n) return;
    float v = fmaxf(deg[i], 1.0f);
    deg[i] = rsqrtf(v);
}

// Pack W [KDIM x wcols] into per-(step,half) float2 fragments, zero-padded:
//   Wpk[(s*2+h)*ldw + n] = ( W[kb][n], W[kb+1][n] ),  kb = 8*(s/2) + 4*h + 2*(s&1)
// Matches the K-permuted A mapping: group g, half h reads X cols [8g+4h..8g+4h+3]
// with one b128 load covering steps 2g (.xy) and 2g+1 (.zw).
__global__ void pack_w(const float* __restrict__ W, v2f* __restrict__ Wpk,
                       int wcols, int ldw) {
    int i = blockIdx.x * blockDim.x + threadIdx.x;
    if (i >= 64 * ldw) return;              // 32 steps * 2 halves
    int sh = i / ldw, n = i - sh * ldw;
    int s = sh >> 1, h = sh & 1;
    int kb = 8 * (s >> 1) + 4 * h + 2 * (s & 1);
    v2f v;
    v.x = (n < wcols) ? W[kb * wcols + n]       : 0.0f;
    v.y = (n < wcols) ? W[(kb + 1) * wcols + n] : 0.0f;
    Wpk[i] = v;
}

// ---------------------------------------------------------------------------
// WMMA f32 GEMM: Y[nrows x YSTRIDE] = diag(rowscale) * X[nrows x 128] * W[128 x *]
// LDW: packed-W column count (= 16 * gridDim.x). YCOLS: real output columns
// (lanes with n >= YCOLS never store, so their columns can hold anything).
// One wave = 64x16 super-tile, 4 independent accumulator chains, b128 A loads,
// one-group software pipeline with peeled last group (branch-free body).
// ---------------------------------------------------------------------------
template <int LDW, int YSTRIDE, int YCOLS, bool HAS_SCALE>
__global__ void gemm_wmma(const float* __restrict__ X,
                          const v2f* __restrict__ Wpk,
                          const float* __restrict__ rowscale,
                          float* __restrict__ Y,
                          int nrows) {
    int lane = threadIdx.x & 31;
    int half = lane >> 4;
    int l    = lane & 15;
    int colbase  = blockIdx.x * 16;
    int rowbase0 = blockIdx.y * 64;
    int n = colbase + l;
    bool nok = (n < YCOLS);

    // B fragment for step s lives at wb[(2*s)*LDW]
    const v2f* wb = Wpk + (size_t)half * LDW + n;

    const float *xr0, *xr1, *xr2, *xr3;
    float s0, s1, s2, s3;
    {
        int m0 = rowbase0 + 0  + l, m1 = rowbase0 + 16 + l;
        int m2 = rowbase0 + 32 + l, m3 = rowbase0 + 48 + l;
        int c0 = (m0 < nrows) ? m0 : 0, c1 = (m1 < nrows) ? m1 : 0;
        int c2 = (m2 < nrows) ? m2 : 0, c3 = (m3 < nrows) ? m3 : 0;
        xr0 = X + (size_t)c0 * KDIM + 4 * half;
        xr1 = X + (size_t)c1 * KDIM + 4 * half;
        xr2 = X + (size_t)c2 * KDIM + 4 * half;
        xr3 = X + (size_t)c3 * KDIM + 4 * half;
        float r0 = HAS_SCALE ? rowscale[c0] : 1.0f;
        float r1 = HAS_SCALE ? rowscale[c1] : 1.0f;
        float r2 = HAS_SCALE ? rowscale[c2] : 1.0f;
        float r3 = HAS_SCALE ? rowscale[c3] : 1.0f;
        s0 = (m0 < nrows) ? r0 : 0.0f; s1 = (m1 < nrows) ? r1 : 0.0f;
        s2 = (m2 < nrows) ? r2 : 0.0f; s3 = (m3 < nrows) ? r3 : 0.0f;
    }

    v8f acc0 = {}, acc1 = {}, acc2 = {}, acc3 = {};

    // Prologue: load group 0
    v4f A0 = *(const v4f*)(xr0);
    v4f A1 = *(const v4f*)(xr1);
    v4f A2 = *(const v4f*)(xr2);
    v4f A3 = *(const v4f*)(xr3);
    v2f B0 = wb[0];
    v2f B1 = wb[2 * LDW];

    #pragma unroll 5
    for (int g = 0; g < 15; ++g) {          // branch-free: g+1 always valid
        v4f a0 = A0, a1 = A1, a2 = A2, a3 = A3;
        v2f b0 = B0, b1 = B1;
        A0 = *(const v4f*)(xr0 + 8 * (g + 1));
        A1 = *(const v4f*)(xr1 + 8 * (g + 1));
        A2 = *(const v4f*)(xr2 + 8 * (g + 1));
        A3 = *(const v4f*)(xr3 + 8 * (g + 1));
        B0 = wb[(4 * (g + 1)) * LDW];
        B1 = wb[(4 * (g + 1) + 2) * LDW];
        acc0 = wmma_step(s0, a0.x, a0.y, b0, acc0);
        acc1 = wmma_step(s1, a1.x, a1.y, b0, acc1);
        acc2 = wmma_step(s2, a2.x, a2.y, b0, acc2);
        acc3 = wmma_step(s3, a3.x, a3.y, b0, acc3);
        acc0 = wmma_step(s0, a0.z, a0.w, b1, acc0);
        acc1 = wmma_step(s1, a1.z, a1.w, b1, acc1);
        acc2 = wmma_step(s2, a2.z, a2.w, b1, acc2);
        acc3 = wmma_step(s3, a3.z, a3.w, b1, acc3);
    }
    // Peeled final group (g = 15)
    acc0 = wmma_step(s0, A0.x, A0.y, B0, acc0);
    acc1 = wmma_step(s1, A1.x, A1.y, B0, acc1);
    acc2 = wmma_step(s2, A2.x, A2.y, B0, acc2);
    acc3 = wmma_step(s3, A3.x, A3.y, B0, acc3);
    acc0 = wmma_step(s0, A0.z, A0.w, B1, acc0);
    acc1 = wmma_step(s1, A1.z, A1.w, B1, acc1);
    acc2 = wmma_step(s2, A2.z, A2.w, B1, acc2);
    acc3 = wmma_step(s3, A3.z, A3.w, B1, acc3);

    if (nok) {
        #pragma unroll
        for (int r = 0; r < 8; ++r) {
            int row = rowbase0 + r + 8 * half;
            if (row < nrows) Y[(size_t)row * YSTRIDE + n] = acc0[r];
        }
        #pragma unroll
        for (int r = 0; r < 8; ++r) {
            int row = rowbase0 + 16 + r + 8 * half;
            if (row < nrows) Y[(size_t)row * YSTRIDE + n] = acc1[r];
        }
        #pragma unroll
        for (int r = 0; r < 8; ++r) {
            int row = rowbase0 + 32 + r + 8 * half;
            if (row < nrows) Y[(size_t)row * YSTRIDE + n] = acc2[r];
        }
        #pragma unroll
        for (int r = 0; r < 8; ++r) {
            int row = rowbase0 + 48 + r + 8 * half;
            if (row < nrows) Y[(size_t)row * YSTRIDE + n] = acc3[r];
        }
    }
}

// ---------------------------------------------------------------------------
// Edge scatter, 128 channels: one wave per edge, float4 gather + 4 atomics/lane
// Both h (51.2MB) and agg (51.2MB) fit in the 192MB L2 -> atomics resolve in L2
// ---------------------------------------------------------------------------
__global__ void scatter128(const long long* __restrict__ ei, int E,
                           const float* __restrict__ h, float* __restrict__ agg) {
    int gid  = blockIdx.x * blockDim.x + threadIdx.x;
    int e    = gid >> 5;
    int lane = gid & 31;
    if (e >= E) return;
    int s = (int)ei[e];
    int d = (int)ei[(size_t)E + e];
    float4 v = *((const float4*)(h + (size_t)s * 128) + lane);
    float* ap = agg + (size_t)d * 128 + lane * 4;
    atomicAdd(ap + 0, v.x);
    atomicAdd(ap + 1, v.y);
    atomicAdd(ap + 2, v.z);
    atomicAdd(ap + 3, v.w);
}

// Edge scatter, 40 tight channels: 10 float4 groups per edge, into d_out
__global__ void scatter40(const long long* __restrict__ ei, int E,
                          const float* __restrict__ h, float* __restrict__ agg) {
    int gid = blockIdx.x * blockDim.x + threadIdx.x;
    int e   = gid / 10;
    int c   = (gid - e * 10) * 4;
    if (e >= E) return;
    int s = (int)ei[e];
    int d = (int)ei[(size_t)E + e];
    float4 v = *((const float4*)(h + (size_t)s * 40 + c));
    float* ap = agg + (size_t)d * 40 + c;
    atomicAdd(ap + 0, v.x);
    atomicAdd(ap + 1, v.y);
    atomicAdd(ap + 2, v.z);
    atomicAdd(ap + 3, v.w);
}

// x2 = relu(agg1 * in_isqrt) * out_isqrt   (in place on agg1 buffer)
__global__ void finalize1(float* __restrict__ a,
                          const float* __restrict__ in_is,
                          const float* __restrict__ out_is, int total) {
    int i = blockIdx.x * blockDim.x + threadIdx.x;
    if (i >= total) return;
    int node = i >> 7; // /128
    float v = fmaxf(a[i] * in_is[node], 0.0f);
    a[i] = v * out_is[node];
}

// out[i,c] *= in_isqrt[i]   (in place on d_out, stride 40)
__global__ void finalize2(float* __restrict__ out,
                          const float* __restrict__ in_is, int N) {
    int i = blockIdx.x * blockDim.x + threadIdx.x;
    int total = N * 40;
    if (i >= total) return;
    int node = i / 40;
    out[i] *= in_is[node];
}

// ---------------------------------------------------------------------------
extern "C" void kernel_launch(void* const* d_in, const int* in_sizes, int n_in,
                              void* d_out, int out_size, void* d_ws, size_t ws_size,
                              hipStream_t stream) {
    const float*     x  = (const float*)d_in[0];      // N x 128
    const long long* ei = (const long long*)d_in[1];  // 2 x E (int64)
    const float*     W1 = (const float*)d_in[2];      // 128 x 128
    const float*     W2 = (const float*)d_in[3];      // 128 x 40
    float*           out = (float*)d_out;             // N x 40

    const int N = in_sizes[0] / 128;
    const int E = in_sizes[1] / 2;

    // Workspace (floats):
    //   [out_isqrt:N][in_isqrt:N][bufH:128N][bufA:128N][Wpk1:64*128*2][Wpk2:64*48*2]
    float* ws     = (float*)d_ws;
    float* out_is = ws;
    float* in_is  = ws + (size_t)N;
    float* bufH   = ws + (size_t)2 * N;                       // h1 proj; later h2 (N x 40)
    float* bufA   = ws + (size_t)2 * N + (size_t)128 * N;     // agg1; later x2
    v2f*   Wpk1   = (v2f*)(ws + (size_t)2 * N + (size_t)256 * N);
    v2f*   Wpk2   = Wpk1 + (size_t)64 * 128;
    float* h2     = bufH;                                     // N x 40 (tight)

    const int TB = 256;

    // --- degrees + weight packing ---
    hipMemsetAsync(out_is, 0, (size_t)2 * N * sizeof(float), stream);
    hipMemsetAsync(bufA,   0, (size_t)128 * N * sizeof(float), stream);
    deg_kernel<<<(E + TB - 1) / TB, TB, 0, stream>>>(ei, E, out_is, in_is);
    isqrt_kernel<<<(2 * N + TB - 1) / TB, TB, 0, stream>>>(out_is, 2 * N);
    pack_w<<<(64 * 128 + TB - 1) / TB, TB, 0, stream>>>(W1, Wpk1, 128, 128);
    pack_w<<<(64 * 48 + TB - 1) / TB, TB, 0, stream>>>(W2, Wpk2, 40, 48);

    // --- layer 1: GEMM (scaled by out_isqrt) -> scatter -> finalize ---
    {
        dim3 grid(128 / 16, (N + 63) / 64);
        gemm_wmma<128, 128, 128, true><<<grid, 32, 0, stream>>>(x, Wpk1, out_is, bufH, N);
    }
    {
        long long tot = (long long)E * 32;
        scatter128<<<(unsigned)((tot + TB - 1) / TB), TB, 0, stream>>>(ei, E, bufH, bufA);
    }
    finalize1<<<(128 * N + TB - 1) / TB, TB, 0, stream>>>(bufA, in_is, out_is, 128 * N);

    // --- layer 2: GEMM (x2 pre-scaled) -> scatter into d_out -> scale in place ---
    hipMemsetAsync(out, 0, (size_t)N * 40 * sizeof(float), stream);
    {
        dim3 grid(48 / 16, (N + 63) / 64);
        gemm_wmma<48, 40, 40, false><<<grid, 32, 0, stream>>>(bufA, Wpk2, nullptr, h2, N);
    }
    {
        long long tot = (long long)E * 10;
        scatter40<<<(unsigned)((tot + TB - 1) / TB), TB, 0, stream>>>(ei, E, h2, out);
    }
    finalize2<<<(N * 40 + TB - 1) / TB, TB, 0, stream>>>(out, in_is, N);
}